// TorchAccelerator_33440615367301
// MI455X (gfx1250) — compile-verified
//
#include <hip/hip_runtime.h>

// Beamline transport: M = A_{N-1} @ ... @ A_0 (6x6 fp32, N = 1e6), then
// out = M @ Sigma @ M^T + noise.
//
// MI455X reasoning:
//  * 432 MFLOP vs 144 MB streamed once -> HBM floor ~6.2 us at 23.3 TB/s.
//    Non-temporal loads; single pass, nothing to block for. The chain loop is
//    issue-bound, so the design minimizes instructions per matrix.
//  * Precision: 1e6 products of (I + 1e-4*G) -> must stay fp32. The only
//    full-fp32 matrix op on CDNA5 is V_WMMA_F32_16X16X4_F32; a 6x6 product
//    (K=6) is two chained K=4 WMMAs, running product resident in the 16x16
//    f32 accumulator.
//  * Row-duplication invariant (kills the ds_bpermutes): the A operand loads
//    rows 8..13 = rows 0..5 (loop-invariant per-lane offsets), so WMMA itself
//    maintains D[8+r] == D[r] bitwise. Lanes 16-31 of P[r] then already hold
//    row r — the B operand rebuild is 4 cndmasks, zero cross-lane ops.
//  * Garbage tolerance: A rows 6,7,14,15 and A K-slots 6,7 are unmasked.
//    They only touch D rows 6,7,14,15 (never consumed/stored); B rows 6,7
//    are kept zero so the A K=6,7 terms vanish. Cols 6..15 stay zero
//    inductively and never mix into cols 0..5.
//  * readfirstlane scalarizes wave/start/cnt -> SALU loop control + SADDR
//    loads instead of exec-mask looping and 64-bit VALU pointer marches.
//  * Matmul is associative -> 3-level tree of ordered chunk products:
//    3907 waves x 256 mats -> 62 waves x 64 partials -> 1 wave finalize.
//
// WMMA f32 16x16x4 VGPR layouts (ISA 7.12.2):
//   A (16x4): lane L<16: {A[L][0],A[L][1]};  lane L>=16: {A[L-16][2],A[L-16][3]}
//   B (4x16): v.x = row K=0 (lanes 0-15) / K=2 (lanes 16-31); v.y = K=1 / K=3
//   C/D (16x16): VGPR r = row r (lanes 0-15), row r+8 (lanes 16-31)

typedef float v2f __attribute__((ext_vector_type(2)));
typedef float v8f __attribute__((ext_vector_type(8)));

__device__ __forceinline__ v8f wmma4(v2f a, v2f b, v8f c) {
  // (neg_a, A, neg_b, B, c_mod, C, reuse_a, reuse_b)
  return __builtin_amdgcn_wmma_f32_16x16x4_f32(false, a, false, b, (short)0, c,
                                               false, false);
}

// P <- mat(6x6, row-major 36 floats) @ P, with P in D-layout under the
// row-duplication invariant (rows 8+r duplicate rows r).
__device__ __forceinline__ void chain_step(const float* __restrict__ mat,
                                           int offLo, int offHi, bool hi,
                                           v8f& P) {
  v2f alo = __builtin_nontemporal_load((const v2f*)(mat + offLo));
  v2f ahi = __builtin_nontemporal_load((const v2f*)(mat + offHi));

  // B rebuild: hi lanes read the duplicated rows already sitting in P[2],P[3].
  v2f blo, bhi;
  blo.x = hi ? P[2] : P[0];
  blo.y = hi ? P[3] : P[1];
  bhi.x = hi ? 0.f : P[4];   // B rows 6,7 MUST stay zero (kills A K=6,7 junk)
  bhi.y = hi ? 0.f : P[5];

  v8f zc = {};
  v8f t = wmma4(ahi, bhi, zc);   // K = 4..5 contribution
  P = wmma4(alo, blo, t);        // + K = 0..3 contribution
}

// Duplicated identity: rows 0..5 and 8..13 both get I6.
__device__ __forceinline__ v8f ident6dup(int lane) {
  const int ml = lane & 15;
  v8f c = {};
  c[0] = (ml == 0) ? 1.f : 0.f;
  c[1] = (ml == 1) ? 1.f : 0.f;
  c[2] = (ml == 2) ? 1.f : 0.f;
  c[3] = (ml == 3) ? 1.f : 0.f;
  c[4] = (ml == 4) ? 1.f : 0.f;
  c[5] = (ml == 5) ? 1.f : 0.f;
  return c;
}

// Per-lane A-operand float offsets implementing row duplication + clamp.
__device__ __forceinline__ void lane_offsets(int lane, int& offLo, int& offHi,
                                             bool& hi) {
  const int ml = lane & 15;
  hi = lane >= 16;
  int m = ml & 7;                 // rows 8..13 re-read rows 0..5
  m = (m < 6) ? m : 0;            // lanes 6,7,14,15: clamp in-bounds (garbage OK)
  offLo = m * 6 + (hi ? 2 : 0);   // K = {0,1} / {2,3}
  offHi = m * 6 + 4;              // K = {4,5} / {6,7-junk, killed by B zeros}
}

// Each wave computes the ordered product of matrices [w*chunk, min(n,(w+1)*chunk))
// (higher index on the left) and stores it as 36 row-major floats at dst[w].
__global__ __launch_bounds__(256)
void beamline_chain_partial(const float* __restrict__ src, int n, int chunk,
                            float* __restrict__ dst) {
  const int lane = threadIdx.x & 31;
  // Force wave-uniform values into SGPRs: scalar loop control + SADDR loads.
  const int wave  = __builtin_amdgcn_readfirstlane(
                      (int)(blockIdx.x * (blockDim.x >> 5) + (threadIdx.x >> 5)));
  const int start = wave * chunk;
  if (start >= n) return;                      // uniform: EXEC stays full
  const int cnt = ((start + chunk) < n ? chunk : (n - start));

  int offLo, offHi; bool hi;
  lane_offsets(lane, offLo, offHi, hi);

  v8f P = ident6dup(lane);
  const float* base = src + (size_t)start * 36;
#pragma unroll 4
  for (int i = 0; i < cnt; ++i)
    chain_step(base + (size_t)i * 36, offLo, offHi, hi, P);

  if (lane < 6) {
#pragma unroll
    for (int r = 0; r < 6; ++r)
      dst[(size_t)wave * 36 + r * 6 + lane] = P[r];  // D layout: (r, lane)
  }
}

// Single wave: fold remaining partials, then out = M @ Sigma @ M^T + noise.
__global__ __launch_bounds__(32)
void beamline_finalize(const float* __restrict__ partials, int n,
                       const float* __restrict__ sigma,
                       const float* __restrict__ noise,
                       float* __restrict__ out) {
  __shared__ float sM[36];
  __shared__ float sS1[36];
  const int  lane = threadIdx.x;
  const int  ml   = lane & 15;
  const bool hiL  = lane >= 16;
  const bool mval = ml < 6;
  const int  m    = mval ? ml : 0;

  int offLo, offHi; bool hi;
  lane_offsets(lane, offLo, offHi, hi);

  v8f P = ident6dup(lane);
  for (int i = 0; i < n; ++i)
    chain_step(partials + (size_t)i * 36, offLo, offHi, hi, P);

  // Park M = P in LDS so it can be re-read in A-layout (D->A is a transposed
  // gather; the LDS round-trip is cheap and in-order within one wave).
  if (lane < 6) {
#pragma unroll
    for (int r = 0; r < 6; ++r) sM[r * 6 + lane] = P[r];
  }
  __syncthreads();

  v8f zc = {};
  // ---- S1 = M @ Sigma ----
  v2f alo, ahi, blo, bhi;
  alo.x = mval ? sM[m * 6 + (hiL ? 2 : 0)] : 0.f;
  alo.y = mval ? sM[m * 6 + (hiL ? 3 : 1)] : 0.f;
  ahi.x = (mval && !hiL) ? sM[m * 6 + 4] : 0.f;
  ahi.y = (mval && !hiL) ? sM[m * 6 + 5] : 0.f;
  blo.x = mval ? sigma[(hiL ? 2 : 0) * 6 + m] : 0.f;   // B[K][N=m]
  blo.y = mval ? sigma[(hiL ? 3 : 1) * 6 + m] : 0.f;
  bhi.x = (mval && !hiL) ? sigma[4 * 6 + m] : 0.f;
  bhi.y = (mval && !hiL) ? sigma[5 * 6 + m] : 0.f;
  v8f S1 = wmma4(alo, blo, wmma4(ahi, bhi, zc));

  if (lane < 6) {
#pragma unroll
    for (int r = 0; r < 6; ++r) sS1[r * 6 + lane] = S1[r];
  }
  __syncthreads();

  // ---- R = S1 @ M^T  (B[K][N] = M[N][K]) ----
  alo.x = mval ? sS1[m * 6 + (hiL ? 2 : 0)] : 0.f;
  alo.y = mval ? sS1[m * 6 + (hiL ? 3 : 1)] : 0.f;
  ahi.x = (mval && !hiL) ? sS1[m * 6 + 4] : 0.f;
  ahi.y = (mval && !hiL) ? sS1[m * 6 + 5] : 0.f;
  blo.x = mval ? sM[m * 6 + (hiL ? 2 : 0)] : 0.f;      // M[N=m][K]
  blo.y = mval ? sM[m * 6 + (hiL ? 3 : 1)] : 0.f;
  bhi.x = (mval && !hiL) ? sM[m * 6 + 4] : 0.f;
  bhi.y = (mval && !hiL) ? sM[m * 6 + 5] : 0.f;
  v8f R = wmma4(alo, blo, wmma4(ahi, bhi, zc));

  if (lane < 6) {
#pragma unroll
    for (int r = 0; r < 6; ++r)
      out[r * 6 + lane] = R[r] + noise[r * 6 + lane];
  }
}

extern "C" void kernel_launch(void* const* d_in, const int* in_sizes, int n_in,
                              void* d_out, int out_size, void* d_ws, size_t ws_size,
                              hipStream_t stream) {
  const float* elems = (const float*)d_in[0];   // (N, 6, 6) fp32
  const float* sigma = (const float*)d_in[1];   // (6, 6)
  const float* noise = (const float*)d_in[2];   // (6, 6)
  float*       out   = (float*)d_out;           // (6, 6)

  const int nMat = in_sizes[0] / 36;            // 1,000,000

  const int C1 = 256;                            // mats per wave, level 1
  const int W1 = (nMat + C1 - 1) / C1;           // 3907 partials (~563 KB in ws)
  const int C2 = 64;                             // partials per wave, level 2
  const int W2 = (W1 + C2 - 1) / C2;             // 62 partials

  float* p1 = (float*)d_ws;
  float* p2 = p1 + (size_t)W1 * 36;

  const int wavesPerBlock = 8;                   // 256 threads = 8 wave32
  dim3 block(256);
  dim3 grid1((W1 + wavesPerBlock - 1) / wavesPerBlock);
  dim3 grid2((W2 + wavesPerBlock - 1) / wavesPerBlock);

  beamline_chain_partial<<<grid1, block, 0, stream>>>(elems, nMat, C1, p1);
  beamline_chain_partial<<<grid2, block, 0, stream>>>(p1, W1, C2, p2);
  beamline_finalize<<<1, dim3(32), 0, stream>>>(p2, W2, sigma, noise, out);
}